// KarateGraph4Att_68599217652369
// MI455X (gfx1250) — compile-verified
//
#include <hip/hip_runtime.h>
#include <hip/hip_bf16.h>

// ---------------------------------------------------------------------------
// 4-layer GAT on gfx1250 (MI455X).
//  - GEMMs: V_WMMA_F32_16X16X4_F32, 64x64 block tile, wave tile 16x32
//    (two accumulators share one A fragment -> 8 WMMA per wave per BK=16).
//  - Attention/aggregation: CSR-by-dst built once per call; one block per
//    destination node does segment max/sum + register-accumulated gather.
//    Zero floating-point atomics; h/out traffic is L2-resident (192MB L2).
// ---------------------------------------------------------------------------

typedef __attribute__((ext_vector_type(2))) float v2f;
typedef __attribute__((ext_vector_type(8))) float v8f;

#define GAT_N   10000
#define GAT_E   160000
#define GAT_EP  (GAT_E + GAT_N)
#define NEG_SLOPE 0.2f
#define GAT_EPS 1e-16f

// ---------------------------------------------------------------------------
// GEMM: H[M x Nd] = A[M x K] * W[K x Nd], fp32 WMMA 16x16x4.
// Block = 256 threads = 8 waves; block tile 64(M) x 64(N); BK = 16.
// Wave w: wm = w & 3 (16-row M subtile), wn = w >> 2 (32-col N subtile).
// ---------------------------------------------------------------------------
__global__ __launch_bounds__(256)
void gemm_wmma_f32(const float* __restrict__ A, const float* __restrict__ W,
                   float* __restrict__ H, int M, int K, int Nd) {
    __shared__ float As[64 * 17];   // 64 rows x 16 cols, pad->17
    __shared__ float Bs[16 * 65];   // 16 rows x 64 cols, pad->65

    const int tid  = threadIdx.x;
    const int lane = tid & 31;
    const int wave = tid >> 5;
    const int wm   = wave & 3;      // 0..3 -> M subtile (16 rows)
    const int wn   = wave >> 2;     // 0..1 -> N subtile (32 cols)

    const int gm0 = blockIdx.x * 64;
    const int gn0 = blockIdx.y * 64;

    // fragment addressing (ISA 32-bit A 16x4 / B 4x16 / C 16x16 layouts)
    const int fr_m  = lane & 15;        // row (A) / col (B,C) within 16
    const int fr_kh = (lane >> 4) * 2;  // lane-half -> K pair {0,1} vs {2,3}
    const int fr_mb = (lane >> 4) * 8;  // C: lanes 16..31 hold rows +8

    // global staging coords
    const int a_row = tid >> 2;         // 0..63
    const int a_c4  = (tid & 3) * 4;    // 0,4,8,12
    const int b_row = tid >> 4;         // 0..15
    const int b_c4  = (tid & 15) * 4;   // 0..60

    v8f acc0 = {};
    v8f acc1 = {};

    for (int k0 = 0; k0 < K; k0 += 16) {
        // ---- stage A tile (64x16) ----
        {
            const int grow = gm0 + a_row;
            float4 av;
            if (grow < M) {
                av = *reinterpret_cast<const float4*>(&A[(size_t)grow * K + k0 + a_c4]);
            } else {
                av = make_float4(0.f, 0.f, 0.f, 0.f);
            }
            float* dst = &As[a_row * 17 + a_c4];
            dst[0] = av.x; dst[1] = av.y; dst[2] = av.z; dst[3] = av.w;
        }
        // ---- stage B tile (16x64) ----
        {
            const float4 bv = *reinterpret_cast<const float4*>(
                &W[(size_t)(k0 + b_row) * Nd + gn0 + b_c4]);
            float* dst = &Bs[b_row * 65 + b_c4];
            dst[0] = bv.x; dst[1] = bv.y; dst[2] = bv.z; dst[3] = bv.w;
        }
        __syncthreads();

        // ---- 4 K-steps x 2 N-subtiles = 8 WMMA ----
        #pragma unroll
        for (int kk = 0; kk < 16; kk += 4) {
            v2f a;
            const int am = (wm * 16 + fr_m) * 17 + kk + fr_kh;
            a.x = As[am];
            a.y = As[am + 1];
            const int bn = wn * 32 + fr_m;
            v2f b0, b1;
            b0.x = Bs[(kk + fr_kh)     * 65 + bn];
            b0.y = Bs[(kk + fr_kh + 1) * 65 + bn];
            b1.x = Bs[(kk + fr_kh)     * 65 + bn + 16];
            b1.y = Bs[(kk + fr_kh + 1) * 65 + bn + 16];
            acc0 = __builtin_amdgcn_wmma_f32_16x16x4_f32(
                false, a, false, b0, (short)0, acc0, false, false);
            acc1 = __builtin_amdgcn_wmma_f32_16x16x4_f32(
                false, a, false, b1, (short)0, acc1, false, false);
        }
        __syncthreads();
    }

    // ---- epilogue ----
    const int col  = gn0 + wn * 32 + fr_m;
    const int row0 = gm0 + wm * 16 + fr_mb;
    #pragma unroll
    for (int r = 0; r < 8; ++r) {
        const int row = row0 + r;
        if (row < M) {
            H[(size_t)row * Nd + col]      = acc0[r];
            H[(size_t)row * Nd + col + 16] = acc1[r];
        }
    }
}

// ---------------------------------------------------------------------------
// Per-node attention scores: ssrc[n] = h[n].a_src ; sdst[n] = h[n].a_dst
// ---------------------------------------------------------------------------
__global__ __launch_bounds__(256)
void scores_kernel(const float* __restrict__ H, const float* __restrict__ asrc,
                   const float* __restrict__ adst, float* __restrict__ ssrc,
                   float* __restrict__ sdst, int D) {
    __shared__ float r1[256], r2[256];
    const int node = blockIdx.x;
    const int tid  = threadIdx.x;
    float p1 = 0.f, p2 = 0.f;
    const float* hrow = H + (size_t)node * D;
    for (int d = tid; d < D; d += 256) {
        const float hv = hrow[d];
        p1 += hv * asrc[d];
        p2 += hv * adst[d];
    }
    r1[tid] = p1; r2[tid] = p2;
    __syncthreads();
    for (int s = 128; s > 0; s >>= 1) {
        if (tid < s) { r1[tid] += r1[tid + s]; r2[tid] += r2[tid + s]; }
        __syncthreads();
    }
    if (tid == 0) { ssrc[node] = r1[0]; sdst[node] = r2[0]; }
}

// ---------------------------------------------------------------------------
__global__ void fill_u32(unsigned* __restrict__ p, unsigned v, int n) {
    const int i = blockIdx.x * blockDim.x + threadIdx.x;
    if (i < n) p[i] = v;
}
__global__ void copy_i32(const int* __restrict__ src, int* __restrict__ dst, int n) {
    const int i = blockIdx.x * blockDim.x + threadIdx.x;
    if (i < n) dst[i] = src[i];
}

// ------------------------- CSR-by-dst construction -------------------------
__device__ __forceinline__ int edge_src(const long long* ei, int e) {
    return (e < GAT_E) ? (int)ei[e] : (e - GAT_E);
}
__device__ __forceinline__ int edge_dst(const long long* ei, int e) {
    return (e < GAT_E) ? (int)ei[GAT_E + e] : (e - GAT_E);
}

__global__ __launch_bounds__(256)
void csr_count(const long long* __restrict__ ei, int* __restrict__ deg) {
    const int e = blockIdx.x * blockDim.x + threadIdx.x;
    if (e >= GAT_EP) return;
    atomicAdd(&deg[edge_dst(ei, e)], 1);
}

// single-block exclusive scan over n<=16K elements -> rowptr[0..n]
__global__ __launch_bounds__(256)
void csr_scan(const int* __restrict__ deg, int* __restrict__ rowptr, int n) {
    __shared__ int sh[256];
    __shared__ int carry;
    const int tid = threadIdx.x;
    if (tid == 0) carry = 0;
    __syncthreads();
    for (int base = 0; base < n; base += 256) {
        const int v = (base + tid < n) ? deg[base + tid] : 0;
        sh[tid] = v;
        __syncthreads();
        for (int off = 1; off < 256; off <<= 1) {
            const int t = (tid >= off) ? sh[tid - off] : 0;
            __syncthreads();
            sh[tid] += t;
            __syncthreads();
        }
        if (base + tid < n) rowptr[base + tid] = carry + sh[tid] - v;
        __syncthreads();
        if (tid == 0) carry += sh[255];
        __syncthreads();
    }
    if (threadIdx.x == 0) rowptr[n] = carry;
}

__global__ __launch_bounds__(256)
void csr_scatter(const long long* __restrict__ ei, int* __restrict__ cursor,
                 int* __restrict__ colidx) {
    const int e = blockIdx.x * blockDim.x + threadIdx.x;
    if (e >= GAT_EP) return;
    const int p = atomicAdd(&cursor[edge_dst(ei, e)], 1);
    colidx[p] = edge_src(ei, e);
}

// sort each node's src list ascending -> deterministic fp accumulation order
__global__ __launch_bounds__(256)
void csr_sort(const int* __restrict__ rowptr, int* __restrict__ colidx, int n) {
    const int i = blockIdx.x * blockDim.x + threadIdx.x;
    if (i >= n) return;
    const int r0 = rowptr[i], r1 = rowptr[i + 1];
    for (int a = r0 + 1; a < r1; ++a) {
        const int key = colidx[a];
        int b = a - 1;
        while (b >= r0 && colidx[b] > key) { colidx[b + 1] = colidx[b]; --b; }
        colidx[b + 1] = key;
    }
}

// ---------------------------------------------------------------------------
// Fused per-dst-node: segment max/sum (softmax) + gather-aggregate + epilogue.
// One 256-thread block per node. NC = ceil(D/256) accumulators per thread.
// mode 0: out = relu(agg + bias); mode 1: out = log_softmax(agg + bias) (D=128)
// ---------------------------------------------------------------------------
template <int NC>
__global__ __launch_bounds__(256)
void gat_node_kernel(const int* __restrict__ rowptr, const int* __restrict__ colidx,
                     const float* __restrict__ ssrc, const float* __restrict__ sdst,
                     const float* __restrict__ H, const float* __restrict__ bias,
                     float* __restrict__ out, int D, int mode) {
    __shared__ float red[256];
    __shared__ float sh_alpha[256];
    __shared__ int   sh_src[256];

    const int n   = blockIdx.x;
    const int tid = threadIdx.x;
    const int r0  = rowptr[n], r1 = rowptr[n + 1];
    const float sd = sdst[n];

    // ---- segment max of LeakyReLU logits ----
    float lm = -3.402823466e+38f;
    for (int j = r0 + tid; j < r1; j += 256) {
        float v = ssrc[colidx[j]] + sd;
        v = (v > 0.f) ? v : NEG_SLOPE * v;
        lm = fmaxf(lm, v);
    }
    red[tid] = lm;
    __syncthreads();
    for (int s = 128; s > 0; s >>= 1) {
        if (tid < s) red[tid] = fmaxf(red[tid], red[tid + s]);
        __syncthreads();
    }
    const float m = red[0];
    __syncthreads();

    // ---- segment sum of exp ----
    float ls = 0.f;
    for (int j = r0 + tid; j < r1; j += 256) {
        float v = ssrc[colidx[j]] + sd;
        v = (v > 0.f) ? v : NEG_SLOPE * v;
        ls += __expf(v - m);
    }
    red[tid] = ls;
    __syncthreads();
    for (int s = 128; s > 0; s >>= 1) {
        if (tid < s) red[tid] += red[tid + s];
        __syncthreads();
    }
    const float inv = 1.f / (red[0] + GAT_EPS);
    __syncthreads();

    // ---- gather-aggregate: acc += alpha * h[src] ----
    float acc[NC];
    #pragma unroll
    for (int i = 0; i < NC; ++i) acc[i] = 0.f;

    for (int base = r0; base < r1; base += 256) {
        const int j = base + tid;
        if (j < r1) {
            const int s = colidx[j];
            float v = ssrc[s] + sd;
            v = (v > 0.f) ? v : NEG_SLOPE * v;
            sh_alpha[tid] = __expf(v - m) * inv;
            sh_src[tid]   = s;
        }
        __syncthreads();
        const int cnt = min(256, r1 - base);
        for (int t = 0; t < cnt; ++t) {
            const float a = sh_alpha[t];
            const float* hr = H + (size_t)sh_src[t] * D;
            #pragma unroll
            for (int i = 0; i < NC; ++i) {
                const int dd = tid + i * 256;
                if (dd < D) acc[i] += a * hr[dd];
            }
        }
        __syncthreads();
    }

    // ---- epilogue ----
    if (mode == 0) {
        #pragma unroll
        for (int i = 0; i < NC; ++i) {
            const int dd = tid + i * 256;
            if (dd < D) {
                const float v = acc[i] + bias[dd];
                out[(size_t)n * D + dd] = v > 0.f ? v : 0.f;
            }
        }
    } else {
        // log_softmax over D == 128 (first 128 threads own one value each)
        const float v = (tid < 128) ? (acc[0] + bias[tid]) : -3.402823466e+38f;
        red[tid] = v;
        __syncthreads();
        for (int s = 128; s > 0; s >>= 1) {
            if (tid < s) red[tid] = fmaxf(red[tid], red[tid + s]);
            __syncthreads();
        }
        const float m2 = red[0];
        __syncthreads();
        red[tid] = (tid < 128) ? __expf(v - m2) : 0.f;
        __syncthreads();
        for (int s = 128; s > 0; s >>= 1) {
            if (tid < s) red[tid] += red[tid + s];
            __syncthreads();
        }
        const float lse = m2 + __logf(red[0]);
        if (tid < 128) out[(size_t)n * 128 + tid] = v - lse;
    }
}

// ---------------------------------------------------------------------------
extern "C" void kernel_launch(void* const* d_in, const int* in_sizes, int n_in,
                              void* d_out, int out_size, void* d_ws, size_t ws_size,
                              hipStream_t stream) {
    (void)in_sizes; (void)n_in; (void)out_size; (void)ws_size;

    const float*     x  = (const float*)d_in[0];
    const long long* ei = (const long long*)d_in[1];
    const float* Wm[4]  = {(const float*)d_in[2],  (const float*)d_in[6],
                           (const float*)d_in[10], (const float*)d_in[14]};
    const float* Asc[4] = {(const float*)d_in[3],  (const float*)d_in[7],
                           (const float*)d_in[11], (const float*)d_in[15]};
    const float* Adc[4] = {(const float*)d_in[4],  (const float*)d_in[8],
                           (const float*)d_in[12], (const float*)d_in[16]};
    const float* Bc[4]  = {(const float*)d_in[5],  (const float*)d_in[9],
                           (const float*)d_in[13], (const float*)d_in[17]};

    const int din [4] = {256, 1024, 1024, 512};
    const int dout[4] = {1024, 1024, 512, 128};
    const int N = GAT_N;

    // workspace carve-up
    float* ws = (float*)d_ws;
    const size_t BIG = (size_t)N * 1024;
    float* h    = ws;                       // [N x 1024]
    float* xb   = h + BIG;                  // [N x 1024]
    float* ssrc = xb + BIG;                 // [N]
    float* sdst = ssrc + 16384;             // [N]
    int* rowptr = (int*)(sdst + 16384);     // [N+1]
    int* cursor = rowptr + 16384;           // [N]
    int* colidx = cursor + 16384;           // [EP]

    // ---- build CSR-by-dst once per call ----
    const int eg = (GAT_EP + 255) / 256;
    const int ng = (N + 255) / 256;
    fill_u32<<<ng, 256, 0, stream>>>((unsigned*)cursor, 0u, N);
    csr_count<<<eg, 256, 0, stream>>>(ei, cursor);
    csr_scan<<<1, 256, 0, stream>>>(cursor, rowptr, N);
    copy_i32<<<ng, 256, 0, stream>>>(rowptr, cursor, N);
    csr_scatter<<<eg, 256, 0, stream>>>(ei, cursor, colidx);
    csr_sort<<<ng, 256, 0, stream>>>(rowptr, colidx, N);

    // ---- 4 GAT layers ----
    const float* xin = x;
    for (int li = 0; li < 4; ++li) {
        const int K = din[li], D = dout[li];

        dim3 ggrid((N + 63) / 64, D / 64);
        gemm_wmma_f32<<<ggrid, 256, 0, stream>>>(xin, Wm[li], h, N, K, D);

        scores_kernel<<<N, 256, 0, stream>>>(h, Asc[li], Adc[li], ssrc, sdst, D);

        if (li < 3) {
            if (D == 1024) {
                gat_node_kernel<4><<<N, 256, 0, stream>>>(rowptr, colidx, ssrc, sdst,
                                                          h, Bc[li], xb, D, 0);
            } else { // D == 512
                gat_node_kernel<2><<<N, 256, 0, stream>>>(rowptr, colidx, ssrc, sdst,
                                                          h, Bc[li], xb, D, 0);
            }
            xin = xb;
        } else {   // D == 128, fused log_softmax
            gat_node_kernel<1><<<N, 256, 0, stream>>>(rowptr, colidx, ssrc, sdst,
                                                      h, Bc[li], (float*)d_out, 128, 1);
        }
    }
}